// GINE_68281390072102
// MI455X (gfx1250) — compile-verified
//
#include <hip/hip_runtime.h>
#include <hip/hip_bf16.h>

typedef __attribute__((ext_vector_type(2))) float v2f;
typedef __attribute__((ext_vector_type(8))) float v8f;
typedef __attribute__((ext_vector_type(4))) unsigned int u32x4;
typedef __attribute__((ext_vector_type(8))) int i32x8;
typedef __attribute__((ext_vector_type(4))) int i32x4;

#define D 128
#define NNODES 100000
#define NEDGES 600000
#define NLAYERS 4
#define NGRAPHS 128
#define NCLASS 10
#define BN_EPS 1e-5f

// ---------------------------------------------------------------- utilities
__global__ void zero_f32(float* p, long long n) {
  long long i = (long long)blockIdx.x * blockDim.x + threadIdx.x;
  if (i < n) p[i] = 0.f;
}

// h[n,d] = emb[x[n], d]
__global__ void embed_kernel(const int* __restrict__ x,
                             const float* __restrict__ emb,
                             float* __restrict__ h, long long total) {
  long long i = (long long)blockIdx.x * blockDim.x + threadIdx.x;
  if (i >= total) return;
  int d = (int)(i & (D - 1));
  long long n = i >> 7;
  h[i] = emb[(long long)x[n] * D + d];
}

// msg = relu(h[src] + edge_attr@ew + eb); agg[dst] += msg
__global__ void edge_msg_kernel(const float* __restrict__ h,
                                const float* __restrict__ ea,
                                const int* __restrict__ src,
                                const int* __restrict__ dst,
                                const float* __restrict__ ew,  // [2,D]
                                const float* __restrict__ eb,  // [D]
                                float* __restrict__ agg, long long total) {
  long long i = (long long)blockIdx.x * blockDim.x + threadIdx.x;
  if (i >= total) return;
  int d = (int)(i & (D - 1));
  long long e = i >> 7;
  float ev = ea[e * 2 + 0] * ew[d] + ea[e * 2 + 1] * ew[D + d] + eb[d];
  float m = h[(long long)src[e] * D + d] + ev;
  m = fmaxf(m, 0.f);
  atomicAdd(&agg[(long long)dst[e] * D + d], m);
}

// ------------------------------------------------- WMMA f32 GEMM, D=128 cols
// Out[M,128] = (A (+A2)) [M,128] @ W[128,128] + bias ; optional ReLU.
// 256-thread block = 8 waves; wave w owns the 16x16 tile (blockIdx.x*16, w*16).
// W is DMA'd once per block into LDS by the Tensor Data Mover; B-fragments
// are then read back with explicit ds_load_b32 (inline asm: the TDM write is
// invisible to the compiler's memory model, so C++-level reads get folded).
// V_WMMA_F32_16X16X4_F32 fragment layout (ISA 7.12.2):
//   A lane l: VGPR j = A[l%16][2*(l/16)+j]
//   B lane l: VGPR j = W[2*(l/16)+j][l%16]
//   C/D lane l: VGPR r = C[8*(l/16)+r][l%16]
__global__ void gemm_wmma_f32(const float* __restrict__ A,
                              const float* __restrict__ A2,  // nullable: fused add
                              const float* __restrict__ W,
                              const float* __restrict__ bias,
                              float* __restrict__ Out, int M, int do_relu) {
  __shared__ float lds_w[D * D];  // 64 KB

  // LDS byte offset of lds_w: low 32 bits of its flat address (the LDS
  // aperture keeps the offset in addr[31:0]). Also keeps lds_w referenced
  // so the group segment stays allocated.
  const unsigned lds_base = (unsigned)(unsigned long long)&lds_w[0];

  // --- TDM: W[128,128] f32 -> LDS. Wave 0 only; tensor ops ignore EXEC, so
  // the other waves must branch around the issue (s_cbranch_execz does that).
  if (threadIdx.x < 32) {
    unsigned long long ga = (unsigned long long)W;
    u32x4 g0;
    g0.x = 1u;                                   // count=1 (valid user D#)
    g0.y = lds_base;                             // lds_addr (bytes)
    g0.z = (unsigned)(ga & 0xffffffffull);       // global_addr[31:0]
    g0.w = (unsigned)((ga >> 32) & 0x1ffffffull) // global_addr[56:32]
           | 0x80000000u;                        // type=2 ("image")
    i32x8 g1;
    g1[0] = 0x00020000;          // workgroup_mask=0, data_size=2 (4 bytes)
    g1[1] = (int)(D << 16);      // tensor_dim0 lo16 (bits 79:48 span w1/w2)
    g1[2] = (int)(D << 16);      // tensor_dim0 hi16=0 | tensor_dim1 lo16
    g1[3] = (int)(D << 16);      // tensor_dim1 hi16=0 | tile_dim0=128
    g1[4] = D;                   // tile_dim1=128 | tile_dim2=0
    g1[5] = D;                   // tensor_dim0_stride[31:0] = 128 elements
    g1[6] = 0;                   // stride0 hi | tensor_dim1_stride lo (2-D: unused)
    g1[7] = 0;
    i32x4 gz = {0, 0, 0, 0};     // groups 2/3: tensor is 2-D
#if defined(__clang_major__) && (__clang_major__ >= 23)
    i32x8 gz8 = {0, 0, 0, 0, 0, 0, 0, 0};
    __builtin_amdgcn_tensor_load_to_lds(g0, g1, gz, gz, gz8, 0);
#else
    __builtin_amdgcn_tensor_load_to_lds(g0, g1, gz, gz, 0);
#endif
    __builtin_amdgcn_s_wait_tensorcnt(0);
  }
  __syncthreads();

  const int wave = threadIdx.x >> 5;
  const int lane = threadIdx.x & 31;
  const int m = lane & 15;
  const int khalf = lane >> 4;  // 0 or 1
  const int row0 = blockIdx.x * 16;
  const int col0 = wave * 16;

  const float* arow = A + (long long)(row0 + m) * D;
  const unsigned baddr0 = lds_base + (unsigned)((col0 + m) * 4);
  v8f acc = {};

  if (A2 != nullptr) {  // uniform SGPR branch: no per-iteration EXEC churn
    const float* arow2 = A2 + (long long)(row0 + m) * D;
#pragma unroll 4
    for (int k = 0; k < D; k += 4) {
      const int ka = k + khalf * 2;
      v2f a, b;
      float bx, by;
      a.x = arow[ka] + arow2[ka];
      a.y = arow[ka + 1] + arow2[ka + 1];
      unsigned addr = baddr0 + (unsigned)(ka * (D * 4));
      // early-clobber outputs: results must not alias the address register
      asm volatile("ds_load_b32 %0, %2\n\t"
                   "ds_load_b32 %1, %2 offset:512\n\t"
                   "s_wait_dscnt 0x0"
                   : "=&v"(bx), "=&v"(by)
                   : "v"(addr));
      b.x = bx;
      b.y = by;
      acc = __builtin_amdgcn_wmma_f32_16x16x4_f32(false, a, false, b, (short)0,
                                                  acc, false, false);
    }
  } else {
#pragma unroll 4
    for (int k = 0; k < D; k += 4) {
      const int ka = k + khalf * 2;
      v2f a, b;
      float bx, by;
      a.x = arow[ka];
      a.y = arow[ka + 1];
      unsigned addr = baddr0 + (unsigned)(ka * (D * 4));
      asm volatile("ds_load_b32 %0, %2\n\t"
                   "ds_load_b32 %1, %2 offset:512\n\t"
                   "s_wait_dscnt 0x0"
                   : "=&v"(bx), "=&v"(by)
                   : "v"(addr));
      b.x = bx;
      b.y = by;
      acc = __builtin_amdgcn_wmma_f32_16x16x4_f32(false, a, false, b, (short)0,
                                                  acc, false, false);
    }
  }

  const float bv = bias ? bias[col0 + m] : 0.f;
#pragma unroll
  for (int r = 0; r < 8; ++r) {
    const int row = row0 + khalf * 8 + r;
    float v = acc[r] + bv;
    if (do_relu) v = fmaxf(v, 0.f);
    Out[(long long)row * D + col0 + m] = v;
  }
}

// --------------------------------------------------------------- batch norm
__global__ void col_stats_kernel(const float* __restrict__ t,
                                 float* __restrict__ sums,
                                 float* __restrict__ sumsq, int M) {
  const int col = threadIdx.x;  // blockDim.x == D
  float s = 0.f, ss = 0.f;
  for (int row = blockIdx.x; row < M; row += gridDim.x) {
    float v = t[(long long)row * D + col];
    s += v;
    ss += v * v;
  }
  atomicAdd(&sums[col], s);
  atomicAdd(&sumsq[col], ss);
}

__global__ void bn_finalize_kernel(const float* __restrict__ sums,
                                   const float* __restrict__ sumsq,
                                   const float* __restrict__ gamma,
                                   const float* __restrict__ beta,
                                   float* __restrict__ scale,
                                   float* __restrict__ shift, float invM) {
  const int c = threadIdx.x;
  float mean = sums[c] * invM;
  float var = sumsq[c] * invM - mean * mean;  // biased variance
  float sc = gamma[c] * rsqrtf(var + BN_EPS);
  scale[c] = sc;
  shift[c] = beta[c] - mean * sc;
}

__global__ void bn_apply_kernel(float* __restrict__ t,
                                const float* __restrict__ scale,
                                const float* __restrict__ shift, int do_relu,
                                long long total) {
  long long i = (long long)blockIdx.x * blockDim.x + threadIdx.x;
  if (i >= total) return;
  const int c = (int)(i & (D - 1));
  float v = t[i] * scale[c] + shift[c];
  if (do_relu) v = fmaxf(v, 0.f);
  t[i] = v;
}

// ------------------------------------------------------------------ pooling
__global__ void pool_accum_kernel(const float* __restrict__ h,
                                  const int* __restrict__ batch,
                                  float* __restrict__ sums,
                                  float* __restrict__ counts, long long total) {
  long long i = (long long)blockIdx.x * blockDim.x + threadIdx.x;
  if (i >= total) return;
  const int d = (int)(i & (D - 1));
  const long long n = i >> 7;
  const int g = batch[n];
  atomicAdd(&sums[(long long)g * D + d], h[i]);
  if (d == 0) atomicAdd(&counts[g], 1.f);
}

__global__ void pool_div_kernel(const float* __restrict__ sums,
                                const float* __restrict__ counts,
                                float* __restrict__ pooled, long long total) {
  long long i = (long long)blockIdx.x * blockDim.x + threadIdx.x;
  if (i >= total) return;
  const int g = (int)(i >> 7);
  pooled[i] = sums[i] / fmaxf(counts[g], 1.f);
}

// final tiny GEMM: out[G,C] = z[G,128] @ cw2[128,C] + cb2
__global__ void cls2_kernel(const float* __restrict__ z,
                            const float* __restrict__ cw2,
                            const float* __restrict__ cb2,
                            float* __restrict__ out) {
  const int g = blockIdx.x;
  const int c = threadIdx.x;
  if (c >= NCLASS) return;
  float acc = cb2[c];
#pragma unroll 8
  for (int k = 0; k < D; ++k) acc += z[(long long)g * D + k] * cw2[k * NCLASS + c];
  out[(long long)g * NCLASS + c] = acc;
}

// ----------------------------------------------------------------- launcher
static inline dim3 g1d(long long n, int b) { return dim3((unsigned)((n + b - 1) / b)); }

extern "C" void kernel_launch(void* const* d_in, const int* in_sizes, int n_in,
                              void* d_out, int out_size, void* d_ws, size_t ws_size,
                              hipStream_t stream) {
  (void)in_sizes; (void)n_in; (void)out_size; (void)ws_size;
  const int* x          = (const int*)d_in[0];
  const int* edge_index = (const int*)d_in[1];
  const float* edge_attr= (const float*)d_in[2];
  const int* batch      = (const int*)d_in[3];
  const float* emb      = (const float*)d_in[4];
  const float* edge_w   = (const float*)d_in[5];   // [L,2,D]
  const float* edge_b   = (const float*)d_in[6];   // [L,D]
  const float* w1       = (const float*)d_in[7];   // [L,D,D]
  const float* b1       = (const float*)d_in[8];
  const float* g1       = (const float*)d_in[9];
  const float* beta1    = (const float*)d_in[10];
  const float* w2       = (const float*)d_in[11];  // [L,D,D]
  const float* b2       = (const float*)d_in[12];
  const float* ng       = (const float*)d_in[13];
  const float* nb       = (const float*)d_in[14];
  const float* cw1      = (const float*)d_in[15];  // [D,D]
  const float* cb1      = (const float*)d_in[16];
  const float* cw2      = (const float*)d_in[17];  // [D,C]
  const float* cb2      = (const float*)d_in[18];
  float* out            = (float*)d_out;

  const int* src = edge_index;            // edge_index[0, :]
  const int* dst = edge_index + NEDGES;   // edge_index[1, :]

  const long long ND = (long long)NNODES * D;   // 12.8M
  const long long ED = (long long)NEDGES * D;   // 76.8M
  const long long GD = (long long)NGRAPHS * D;

  // workspace carve-up (floats)
  float* ws = (float*)d_ws;
  float* h     = ws;              ws += ND;
  float* agg   = ws;              ws += ND;
  float* t     = ws;              ws += ND;
  float* sums  = ws;              ws += D;
  float* sumsq = ws;              ws += D;
  float* scale = ws;              ws += D;
  float* shift = ws;              ws += D;
  float* psum  = ws;              ws += GD;
  float* cnt   = ws;              ws += NGRAPHS;
  float* pooled= ws;              ws += GD;
  float* cls1  = ws;              ws += GD;

  const int B = 256;

  // h = emb[x]
  embed_kernel<<<g1d(ND, B), B, 0, stream>>>(x, emb, h, ND);

  for (int l = 0; l < NLAYERS; ++l) {
    const float* ew_l = edge_w + (long long)l * 2 * D;
    const float* eb_l = edge_b + (long long)l * D;
    const float* w1_l = w1 + (long long)l * D * D;
    const float* b1_l = b1 + (long long)l * D;
    const float* g1_l = g1 + (long long)l * D;
    const float* be_l = beta1 + (long long)l * D;
    const float* w2_l = w2 + (long long)l * D * D;
    const float* b2_l = b2 + (long long)l * D;
    const float* ng_l = ng + (long long)l * D;
    const float* nb_l = nb + (long long)l * D;

    // agg = scatter_sum(relu(h[src] + edge_attr@ew + eb), dst)
    zero_f32<<<g1d(ND, B), B, 0, stream>>>(agg, ND);
    edge_msg_kernel<<<g1d(ED, B), B, 0, stream>>>(h, edge_attr, src, dst, ew_l,
                                                  eb_l, agg, ED);

    // t = (h + agg) @ w1 + b1   (WMMA; A-operand fused add; W via TDM->LDS)
    gemm_wmma_f32<<<NNODES / 16, 256, 0, stream>>>(h, agg, w1_l, b1_l, t,
                                                   NNODES, /*relu=*/0);
    // t = relu(BN(t; g1, beta1))
    zero_f32<<<1, 2 * D, 0, stream>>>(sums, 2 * D);  // sums+sumsq contiguous
    col_stats_kernel<<<512, D, 0, stream>>>(t, sums, sumsq, NNODES);
    bn_finalize_kernel<<<1, D, 0, stream>>>(sums, sumsq, g1_l, be_l, scale,
                                            shift, 1.f / NNODES);
    bn_apply_kernel<<<g1d(ND, B), B, 0, stream>>>(t, scale, shift, 1, ND);

    // h = BN(t @ w2 + b2; ng, nb) [, relu if not last layer]
    gemm_wmma_f32<<<NNODES / 16, 256, 0, stream>>>(t, nullptr, w2_l, b2_l, h,
                                                   NNODES, /*relu=*/0);
    zero_f32<<<1, 2 * D, 0, stream>>>(sums, 2 * D);
    col_stats_kernel<<<512, D, 0, stream>>>(h, sums, sumsq, NNODES);
    bn_finalize_kernel<<<1, D, 0, stream>>>(sums, sumsq, ng_l, nb_l, scale,
                                            shift, 1.f / NNODES);
    bn_apply_kernel<<<g1d(ND, B), B, 0, stream>>>(h, scale, shift,
                                                  (l != NLAYERS - 1) ? 1 : 0, ND);
  }

  // global mean pool
  zero_f32<<<g1d(GD + NGRAPHS, B), B, 0, stream>>>(psum, GD + NGRAPHS);  // psum+cnt contiguous
  pool_accum_kernel<<<g1d(ND, B), B, 0, stream>>>(h, batch, psum, cnt, ND);
  pool_div_kernel<<<g1d(GD, B), B, 0, stream>>>(psum, cnt, pooled, GD);

  // classifier: relu(pooled @ cw1 + cb1) @ cw2 + cb2
  gemm_wmma_f32<<<NGRAPHS / 16, 256, 0, stream>>>(pooled, nullptr, cw1, cb1,
                                                  cls1, NGRAPHS, /*relu=*/1);
  cls2_kernel<<<NGRAPHS, 32, 0, stream>>>(cls1, cw2, cb2, out);
}